// SparseMMoE_36721970381429
// MI455X (gfx1250) — compile-verified
//
#include <hip/hip_runtime.h>
#include <math.h>

#define BN   32768
#define DN   512
#define ON   16
#define EN   16
#define TN   3
#define H1N  16
#define H2N  8
#define KSEL 4

#define NCOL 304      // 256 expert cols (16e x 16h) + 48 gate cols (3t x 16e)
#define NTILE 19      // 16 expert tiles + 3 gate tiles
#define XS_STRIDE 68  // stride % 64 == 4 -> conflict-free paired reads

typedef __attribute__((ext_vector_type(2))) float v2f;
typedef __attribute__((ext_vector_type(8))) float v8f;
typedef int v4i __attribute__((vector_size(16)));   // matches builtin param type

#if __has_builtin(__builtin_amdgcn_global_load_async_to_lds_b32) && \
    __has_builtin(__builtin_amdgcn_global_load_async_to_lds_b128)
#define HAVE_ASYNC_LDS 1
#else
#define HAVE_ASYNC_LDS 0
#endif

#if HAVE_ASYNC_LDS
__device__ __forceinline__ void async_b32(const float* g, float* l) {
  __builtin_amdgcn_global_load_async_to_lds_b32(
      (__attribute__((address_space(1))) int*)g,
      (__attribute__((address_space(3))) int*)l, 0, 0);
}
__device__ __forceinline__ void async_b128(const float* g, float* l) {
  __builtin_amdgcn_global_load_async_to_lds_b128(
      (__attribute__((address_space(1))) v4i*)g,
      (__attribute__((address_space(3))) v4i*)l, 0, 0);
}
#endif

__device__ __forceinline__ void wait_async_lds() {
#if HAVE_ASYNC_LDS
#if __has_builtin(__builtin_amdgcn_s_wait_asynccnt)
  __builtin_amdgcn_s_wait_asynccnt(0);
#else
  asm volatile("s_wait_asynccnt 0x0" ::: "memory");
#endif
#endif
}

// --------------------------------------------------------------------------
// Kernel 1: fp32 WMMA GEMM.
//   h1[b, e*16+h]  = relu(x[b,:] @ W1[e,:,h] + b1[e,h])     (256 cols)
//   logits[t,b,e]  =       x[b,:] @ wg[t,:,e] + bg[t,e]     ( 48 cols)
// One wave owns a 16-row tile and all 19 column tiles (A reused 19x).
// Weights staged per K-chunk into LDS as float2 K-pairs, [kpair][col], so a
// lane's B fragment for tile j is a ds_load_b64 at constant offset j*128B
// from one base (2addr-mergeable, bank-conflict-free). All 19 fragments are
// loaded into distinct registers BEFORE the 19 WMMAs so dscnt waits stage.
// --------------------------------------------------------------------------
__global__ __launch_bounds__(256) void k1_gemm(
    const float* __restrict__ x,  const float* __restrict__ W1,
    const float* __restrict__ b1, const float* __restrict__ wg,
    const float* __restrict__ bg, float* __restrict__ h1,
    float* __restrict__ logits)
{
  __shared__ float xs_all[8 * 16 * XS_STRIDE];   // 34,816 B
  __shared__ float wl2[32 * NCOL * 2];           // 77,824 B  (float2 [p][col])
  const int tid  = threadIdx.x;
  const int wave = tid >> 5;
  const int lane = tid & 31;
  const int m    = lane & 15;       // row (A) / col (B,C,D) within tile
  const int half = lane >> 4;       // lane group 0/1
  const int row_base = (blockIdx.x * 8 + wave) * 16;
  float* xs = xs_all + wave * (16 * XS_STRIDE);

  v8f acc[NTILE];                   // 16 expert tiles + 3 gate tiles
  #pragma unroll
  for (int j = 0; j < NTILE; ++j)
    #pragma unroll
    for (int i = 0; i < 8; ++i) acc[j][i] = 0.0f;

  for (int kc = 0; kc < DN; kc += 64) {
    __syncthreads();   // protect LDS against WAR from previous chunk

    // ---- stage x tile: 16 rows x 64 cols, 16B per lane per step ----
    #pragma unroll
    for (int j = 0; j < 8; ++j) {
      int q = j * 32 + lane;            // quad index 0..255
      int r = q >> 4;                   // 16 quads per row
      int c = (q & 15) * 4;
#if HAVE_ASYNC_LDS
      async_b128(&x[(size_t)(row_base + r) * DN + kc + c], &xs[r * XS_STRIDE + c]);
#else
      *(float4*)&xs[r * XS_STRIDE + c] =
          *(const float4*)&x[(size_t)(row_base + r) * DN + kc + c];
#endif
    }

    // ---- stage weight chunk as K-pairs: wl2[p][col] = {W[2p][col], W[2p+1][col]} ----
    for (int p = 0; p < 32; ++p) {
      const int k0 = kc + 2 * p;
      for (int col = tid; col < NCOL; col += 256) {
        const float* s0 = (col < 256)
            ? &W1[(size_t)(col >> 4) * (DN * H1N) + (size_t)k0 * H1N + (col & 15)]
            : &wg[(size_t)((col - 256) >> 4) * (DN * EN) + (size_t)k0 * EN + ((col - 256) & 15)];
        const float* s1 = s0 + ((col < 256) ? H1N : EN);   // next k row
        float* dst = &wl2[(p * NCOL + col) * 2];
#if HAVE_ASYNC_LDS
        async_b32(s0, dst);
        async_b32(s1, dst + 1);
#else
        dst[0] = s0[0];
        dst[1] = s1[0];
#endif
      }
    }
    wait_async_lds();
    __syncthreads();

    // ---- hot loop: batched DS loads then 19 WMMA per k-step ----
    for (int kk = 0; kk < 64; kk += 4) {
      const int kl = kk + 2 * half;
      const int p  = (kk >> 1) + half;           // K-pair index for this lane
      const v2f a = *(const v2f*)&xs[m * XS_STRIDE + kl];
      const float* wbase = &wl2[(p * NCOL + m) * 2];
      v2f bfr[NTILE];
      #pragma unroll
      for (int j = 0; j < NTILE; ++j)            // tile j at byte offset j*128
        bfr[j] = *(const v2f*)(wbase + j * 32);
      #pragma unroll
      for (int j = 0; j < NTILE; ++j)
        acc[j] = __builtin_amdgcn_wmma_f32_16x16x4_f32(
            false, a, false, bfr[j], (short)0, acc[j], false, false);
    }
  }

  // Epilogue. D layout: VGPR r holds (M = r + 8*half, N = m).
  #pragma unroll
  for (int r = 0; r < 8; ++r) {
    const int row = row_base + r + 8 * half;
    #pragma unroll
    for (int e = 0; e < EN; ++e)
      h1[(size_t)row * (EN * H1N) + e * H1N + m] =
          fmaxf(acc[e][r] + b1[e * H1N + m], 0.0f);
    #pragma unroll
    for (int t = 0; t < TN; ++t)
      logits[(size_t)t * BN * EN + (size_t)row * EN + m] =
          acc[16 + t][r] + bg[t * EN + m];
  }
}

// --------------------------------------------------------------------------
// Kernel 2: per-row top-4 gating + sparse expert tail + combine.
// --------------------------------------------------------------------------
__global__ __launch_bounds__(256) void k2_experts(
    const float* __restrict__ logits, const float* __restrict__ h1,
    const float* __restrict__ W2, const float* __restrict__ b2,
    const float* __restrict__ W3, const float* __restrict__ b3,
    float* __restrict__ out, float* __restrict__ g_imp,
    float* __restrict__ g_cnt)
{
  __shared__ float W2s[EN * H1N * H2N];   // 2048
  __shared__ float W3s[EN * H2N * ON];    // 2048
  __shared__ float b2s[EN * H2N];
  __shared__ float b3s[EN * ON];
  __shared__ float impS[TN * EN];
  __shared__ float cntS[TN * EN];
  const int tid = threadIdx.x;
  for (int i = tid; i < EN * H1N * H2N; i += 256) W2s[i] = W2[i];
  for (int i = tid; i < EN * H2N * ON; i += 256) W3s[i] = W3[i];
  for (int i = tid; i < EN * H2N; i += 256) b2s[i] = b2[i];
  for (int i = tid; i < EN * ON; i += 256) b3s[i] = b3[i];
  if (tid < TN * EN) { impS[tid] = 0.0f; cntS[tid] = 0.0f; }
  __syncthreads();

  const int b = blockIdx.x * 256 + tid;
  int   ti[TN][KSEL];
  float tg[TN][KSEL];
  #pragma unroll
  for (int t = 0; t < TN; ++t) {
    float lg[EN];
    #pragma unroll
    for (int e = 0; e < EN; ++e)
      lg[e] = logits[(size_t)t * BN * EN + (size_t)b * EN + e];
    unsigned used = 0;
    #pragma unroll
    for (int k = 0; k < KSEL; ++k) {    // top-4, first-index wins ties
      float best = -3.4e38f; int bi = 0;
      #pragma unroll
      for (int e = 0; e < EN; ++e) {
        bool ok = !((used >> e) & 1u);
        if (ok && lg[e] > best) { best = lg[e]; bi = e; }
      }
      used |= (1u << bi);
      ti[t][k] = bi; tg[t][k] = best;
    }
    const float m0 = tg[t][0];
    float s = 0.0f;
    #pragma unroll
    for (int k = 0; k < KSEL; ++k) { tg[t][k] = expf(tg[t][k] - m0); s += tg[t][k]; }
    const float inv = 1.0f / s;
    #pragma unroll
    for (int k = 0; k < KSEL; ++k) {
      tg[t][k] *= inv;
      atomicAdd(&impS[t * EN + ti[t][k]], tg[t][k]);
      atomicAdd(&cntS[t * EN + ti[t][k]], 1.0f);
    }
  }

  float y[TN][ON];
  #pragma unroll
  for (int t = 0; t < TN; ++t)
    #pragma unroll
    for (int o = 0; o < ON; ++o) y[t][o] = 0.0f;

  const float* h1row = h1 + (size_t)b * (EN * H1N);
  for (int e = 0; e < EN; ++e) {
    float g0 = 0.0f, g1 = 0.0f, g2 = 0.0f;
    #pragma unroll
    for (int k = 0; k < KSEL; ++k) {
      if (ti[0][k] == e) g0 = tg[0][k];
      if (ti[1][k] == e) g1 = tg[1][k];
      if (ti[2][k] == e) g2 = tg[2][k];
    }
    if (g0 != 0.0f || g1 != 0.0f || g2 != 0.0f) {
      float hr[H1N];
      #pragma unroll
      for (int h = 0; h < H1N; ++h) hr[h] = h1row[e * H1N + h];
      float h2v[H2N];
      #pragma unroll
      for (int j = 0; j < H2N; ++j) {
        float a = b2s[e * H2N + j];
        #pragma unroll
        for (int h = 0; h < H1N; ++h) a += hr[h] * W2s[e * (H1N * H2N) + h * H2N + j];
        h2v[j] = fmaxf(a, 0.0f);
      }
      float eo[ON];
      #pragma unroll
      for (int o = 0; o < ON; ++o) {
        float a = b3s[e * ON + o];
        #pragma unroll
        for (int j = 0; j < H2N; ++j) a += h2v[j] * W3s[e * (H2N * ON) + j * ON + o];
        eo[o] = fmaxf(a, 0.0f);
      }
      #pragma unroll
      for (int o = 0; o < ON; ++o) {
        y[0][o] += g0 * eo[o];
        y[1][o] += g1 * eo[o];
        y[2][o] += g2 * eo[o];
      }
    }
  }
  #pragma unroll
  for (int t = 0; t < TN; ++t)
    #pragma unroll
    for (int o = 0; o < ON; ++o)
      out[(size_t)t * BN * ON + (size_t)b * ON + o] = y[t][o];

  __syncthreads();
  if (tid < TN * EN) {
    atomicAdd(&g_imp[tid], impS[tid]);
    atomicAdd(&g_cnt[tid], cntS[tid]);
  }
}

// --------------------------------------------------------------------------
// Kernel 3: sum_b logsumexp(x[b,:]) via one wave32 per row.
// --------------------------------------------------------------------------
__global__ __launch_bounds__(256) void k3_rz(const float* __restrict__ x,
                                             float* __restrict__ rz)
{
  const int wave = threadIdx.x >> 5;
  const int lane = threadIdx.x & 31;
  const int row  = blockIdx.x * 8 + wave;
  const float* xr = x + (size_t)row * DN;
  float mx = -3.4e38f;
  #pragma unroll
  for (int j = 0; j < DN / 32; ++j) mx = fmaxf(mx, xr[lane + 32 * j]);
  #pragma unroll
  for (int off = 16; off > 0; off >>= 1) mx = fmaxf(mx, __shfl_xor(mx, off, 32));
  float s = 0.0f;
  #pragma unroll
  for (int j = 0; j < DN / 32; ++j) s += expf(xr[lane + 32 * j] - mx);
  #pragma unroll
  for (int off = 16; off > 0; off >>= 1) s += __shfl_xor(s, off, 32);
  if (lane == 0) atomicAdd(rz, mx + logf(s));
}

// --------------------------------------------------------------------------
// Kernel 4: scalar losses.
// --------------------------------------------------------------------------
__global__ void k4_losses(const float* __restrict__ imp,
                          const float* __restrict__ cnt,
                          const float* __restrict__ rz,
                          float* __restrict__ out_scalars)
{
  if (threadIdx.x == 0 && blockIdx.x == 0) {
    float lb = 0.0f;
    for (int t = 0; t < TN; ++t) {
      const float* iv = imp + t * EN;
      const float* lv = cnt + t * EN;
      float mi = 0.0f, ml = 0.0f;
      for (int e = 0; e < EN; ++e) { mi += iv[e]; ml += lv[e]; }
      mi *= (1.0f / EN); ml *= (1.0f / EN);
      float vi = 0.0f, vl = 0.0f;
      for (int e = 0; e < EN; ++e) {
        float di = iv[e] - mi, dl = lv[e] - ml;
        vi += di * di; vl += dl * dl;
      }
      vi *= (1.0f / (EN - 1)); vl *= (1.0f / (EN - 1));   // ddof=1
      float cvi = vi / (mi * mi + 1e-10f);
      float cvl = vl / (ml * ml + 1e-10f);
      float olmo = 0.0f;
      for (int e = 0; e < EN; ++e) olmo += iv[e] * (lv[e] / (float)BN);
      olmo *= (float)EN;
      lb += (cvi + cvl) * 0.01f + olmo * 0.01f;
    }
    out_scalars[0] = lb;
    out_scalars[1] = (float)TN * (rz[0] / (float)BN) * 0.001f;
  }
}

// --------------------------------------------------------------------------
extern "C" void kernel_launch(void* const* d_in, const int* in_sizes, int n_in,
                              void* d_out, int out_size, void* d_ws, size_t ws_size,
                              hipStream_t stream) {
  const float* x  = (const float*)d_in[0];
  const float* wg = (const float*)d_in[1];
  const float* bg = (const float*)d_in[2];
  const float* W1 = (const float*)d_in[3];
  const float* b1 = (const float*)d_in[4];
  const float* W2 = (const float*)d_in[5];
  const float* b2 = (const float*)d_in[6];
  const float* W3 = (const float*)d_in[7];
  const float* b3 = (const float*)d_in[8];
  float* out = (float*)d_out;

  const size_t h1_elems = (size_t)BN * EN * H1N;          // 8.4M floats
  const size_t lg_elems = (size_t)TN * BN * EN;           // 1.6M floats
  const size_t need = (h1_elems + lg_elems + 2 * TN * EN + 1) * sizeof(float);
  if (ws_size < need) return;

  float* h1s    = (float*)d_ws;
  float* logits = h1s + h1_elems;
  float* imp    = logits + lg_elems;
  float* cnt    = imp + TN * EN;
  float* rz     = cnt + TN * EN;

  (void)hipMemsetAsync(imp, 0, (2 * TN * EN + 1) * sizeof(float), stream);

  k1_gemm   <<<BN / 128, 256, 0, stream>>>(x, W1, b1, wg, bg, h1s, logits);
  k2_experts<<<BN / 256, 256, 0, stream>>>(logits, h1s, W2, b2, W3, b3, out, imp, cnt);
  k3_rz     <<<BN / 8,   256, 0, stream>>>(x, rz);
  k4_losses <<<1, 32, 0, stream>>>(imp, cnt, rz, out + (size_t)TN * BN * ON);
}